// GAT_18116172055006
// MI455X (gfx1250) — compile-verified
//
#include <hip/hip_runtime.h>

// ---------------- problem constants (from reference) ----------------
#define NN   100000
#define EE   1600000
#define ETOT (EE + NN)      // edges + self loops
#define DIN  128
#define SLOPE 0.2f
#define NEG_HUGE (-3.402823466e38f)

typedef __attribute__((ext_vector_type(2))) float v2f;
typedef __attribute__((ext_vector_type(8))) float v8f;

__device__ __forceinline__ float lrelu(float v) { return v > 0.f ? v : SLOPE * v; }

// float atomic max via signed-max / unsigned-min trick (valid for all-finite data)
__device__ __forceinline__ void atomicMaxF(float* addr, float val) {
  if (val >= 0.f) atomicMax((int*)addr, __float_as_int(val));
  else            atomicMin((unsigned int*)addr, __float_as_uint(val));
}

__device__ __forceinline__ void edge_sd(const int* __restrict__ ei, int e, int& src, int& dst) {
  if (e < EE) { src = ei[e]; dst = ei[EE + e]; }
  else        { src = dst = e - EE; }          // self loop
}

// ---------------- utility fill ----------------
__global__ void k_fill(float* __restrict__ p, float v, int n) {
  int i = blockIdx.x * blockDim.x + threadIdx.x;
  if (i < n) p[i] = v;
}

// ---------------- layer 1 GEMM: h[N,64] = x[N,128] @ W1[128,64], fp32 WMMA ----------------
// block = 128 threads = 4 waves; wave w owns the 16-col tile [16w,16w+16); block owns 16 rows.
__global__ void k_gemm1(const float* __restrict__ x, const float* __restrict__ W,
                        float* __restrict__ h) {
  const int rowBase = blockIdx.x * 16;
  const int lane    = threadIdx.x & 31;
  const int colBase = (threadIdx.x >> 5) * 16;
  const int m    = lane & 15;
  const int half = lane >> 4;
  const float* xr = x + (size_t)(rowBase + m) * DIN;
  v8f c = {};
  #pragma unroll 4
  for (int k0 = 0; k0 < DIN; k0 += 4) {
    const int k = k0 + 2 * half;
    v2f a, b;
    a[0] = xr[k];
    a[1] = xr[k + 1];
    b[0] = W[(size_t)k * 64 + colBase + m];
    b[1] = W[(size_t)(k + 1) * 64 + colBase + m];
    c = __builtin_amdgcn_wmma_f32_16x16x4_f32(false, a, false, b, (short)0, c, false, false);
  }
  #pragma unroll
  for (int v = 0; v < 8; ++v)
    h[(size_t)(rowBase + v + 8 * half) * 64 + colBase + m] = c[v];
}

// ---------------- layer 1 attention logits per node: a_s,a_d [N,2] ----------------
__global__ void k_attn1(const float* __restrict__ h, const float* __restrict__ asrc,
                        const float* __restrict__ adst, float* __restrict__ as_,
                        float* __restrict__ ad_) {
  int idx = blockIdx.x * blockDim.x + threadIdx.x;
  if (idx >= NN * 2) return;
  const int n = idx >> 1, hh = idx & 1;
  const float* hp = h + (size_t)n * 64 + hh * 32;
  const float* sp = asrc + hh * 32;
  const float* dp = adst + hh * 32;
  float sa = 0.f, sd = 0.f;
  #pragma unroll
  for (int f = 0; f < 32; ++f) { float v = hp[f]; sa += v * sp[f]; sd += v * dp[f]; }
  as_[idx] = sa;
  ad_[idx] = sd;
}

// ---------------- layer 1 edge passes ----------------
__global__ void k_edge1_max(const int* __restrict__ ei, const float* __restrict__ as_,
                            const float* __restrict__ ad_, float* __restrict__ mx) {
  int e = blockIdx.x * blockDim.x + threadIdx.x;
  if (e >= ETOT) return;
  int src, dst; edge_sd(ei, e, src, dst);
  #pragma unroll
  for (int hh = 0; hh < 2; ++hh) {
    float v = lrelu(as_[src * 2 + hh] + ad_[dst * 2 + hh]);
    atomicMaxF(&mx[dst * 2 + hh], v);
  }
}

__global__ void k_edge1_z(const int* __restrict__ ei, const float* __restrict__ as_,
                          const float* __restrict__ ad_, const float* __restrict__ mx,
                          float* __restrict__ s, float* __restrict__ z) {
  int e = blockIdx.x * blockDim.x + threadIdx.x;
  if (e >= ETOT) return;
  int src, dst; edge_sd(ei, e, src, dst);
  #pragma unroll
  for (int hh = 0; hh < 2; ++hh) {
    float v  = lrelu(as_[src * 2 + hh] + ad_[dst * 2 + hh]);
    float zz = expf(v - mx[dst * 2 + hh]);
    z[(size_t)e * 2 + hh] = zz;
    atomicAdd(&s[dst * 2 + hh], zz);
  }
}

// ---------------- shared scatter: acc[dst,c] += alpha * feat[src,c]  (c = head*32+f) ----------------
// thread-per-(edge,channel): 64 consecutive threads share an edge -> coalesced loads + atomics.
__global__ void k_scatter(const int* __restrict__ ei, const float* __restrict__ feat,
                          const float* __restrict__ z, const float* __restrict__ s,
                          float* __restrict__ acc) {
  unsigned tid = blockIdx.x * blockDim.x + threadIdx.x;
  if (tid >= (unsigned)ETOT * 64u) return;
  const int e  = (int)(tid >> 6);
  const int c  = (int)(tid & 63u);
  const int hh = c >> 5;
  int src, dst; edge_sd(ei, e, src, dst);
  float alpha = z[(size_t)e * 2 + hh] / (s[dst * 2 + hh] + 1e-16f);
  atomicAdd(&acc[(size_t)dst * 64 + c], alpha * feat[(size_t)src * 64 + c]);
}

// ---------------- layer 1 node epilogue: mean heads + bias + relu ----------------
__global__ void k_node1(const float* __restrict__ acc, const float* __restrict__ b1,
                        float* __restrict__ h1) {
  int idx = blockIdx.x * blockDim.x + threadIdx.x;
  if (idx >= NN * 32) return;
  const int n = idx >> 5, f = idx & 31;
  float v = 0.5f * (acc[(size_t)n * 64 + f] + acc[(size_t)n * 64 + 32 + f]) + b1[f];
  h1[idx] = v > 0.f ? v : 0.f;
}

// ---------------- layer 2 GEMM: hl/hr[N,64] = h1[N,32] @ Wl/Wr + bl/br, shared A frag ----------------
__global__ void k_gemm2(const float* __restrict__ h1, const float* __restrict__ Wl,
                        const float* __restrict__ bl, const float* __restrict__ Wr,
                        const float* __restrict__ br, float* __restrict__ hl,
                        float* __restrict__ hr) {
  const int rowBase = blockIdx.x * 16;
  const int lane    = threadIdx.x & 31;
  const int colBase = (threadIdx.x >> 5) * 16;
  const int m    = lane & 15;
  const int half = lane >> 4;
  const float* ar = h1 + (size_t)(rowBase + m) * 32;
  v8f cl = {}, cr = {};
  #pragma unroll
  for (int k0 = 0; k0 < 32; k0 += 4) {
    const int k = k0 + 2 * half;
    v2f a, bL, bR;
    a[0] = ar[k];
    a[1] = ar[k + 1];
    bL[0] = Wl[(size_t)k * 64 + colBase + m];
    bL[1] = Wl[(size_t)(k + 1) * 64 + colBase + m];
    bR[0] = Wr[(size_t)k * 64 + colBase + m];
    bR[1] = Wr[(size_t)(k + 1) * 64 + colBase + m];
    cl = __builtin_amdgcn_wmma_f32_16x16x4_f32(false, a, false, bL, (short)0, cl, false, false);
    cr = __builtin_amdgcn_wmma_f32_16x16x4_f32(false, a, false, bR, (short)0, cr, false, false);
  }
  const float bli = bl[colBase + m], bri = br[colBase + m];
  #pragma unroll
  for (int v = 0; v < 8; ++v) {
    const size_t row = (size_t)(rowBase + v + 8 * half);
    hl[row * 64 + colBase + m] = cl[v] + bli;
    hr[row * 64 + colBase + m] = cr[v] + bri;
  }
}

// ---------------- layer 2 edge logits: wave-per-edge dot(lrelu(hl[src]+hr[dst]), att2) ----------------
__global__ void k_edge2_logits(const int* __restrict__ ei, const float* __restrict__ hl,
                               const float* __restrict__ hr, const float* __restrict__ att,
                               float* __restrict__ e2, float* __restrict__ mx) {
  const int w    = blockIdx.x * (blockDim.x >> 5) + (threadIdx.x >> 5);
  const int lane = threadIdx.x & 31;
  if (w >= ETOT) return;
  int src, dst; edge_sd(ei, w, src, dst);
  float p0 = lrelu(hl[(size_t)src * 64 + lane]      + hr[(size_t)dst * 64 + lane])      * att[lane];
  float p1 = lrelu(hl[(size_t)src * 64 + 32 + lane] + hr[(size_t)dst * 64 + 32 + lane]) * att[32 + lane];
  #pragma unroll
  for (int off = 16; off > 0; off >>= 1) {
    p0 += __shfl_xor(p0, off, 32);
    p1 += __shfl_xor(p1, off, 32);
  }
  if (lane == 0) {
    e2[(size_t)w * 2 + 0] = p0;
    e2[(size_t)w * 2 + 1] = p1;
    atomicMaxF(&mx[dst * 2 + 0], p0);
    atomicMaxF(&mx[dst * 2 + 1], p1);
  }
}

__global__ void k_edge2_z(const int* __restrict__ ei, float* __restrict__ z /* in e2, out z2 */,
                          const float* __restrict__ mx, float* __restrict__ s) {
  int e = blockIdx.x * blockDim.x + threadIdx.x;
  if (e >= ETOT) return;
  int src, dst; edge_sd(ei, e, src, dst);
  (void)src;
  #pragma unroll
  for (int hh = 0; hh < 2; ++hh) {
    float zz = expf(z[(size_t)e * 2 + hh] - mx[dst * 2 + hh]);
    z[(size_t)e * 2 + hh] = zz;
    atomicAdd(&s[dst * 2 + hh], zz);
  }
}

// ---------------- layer 2 node epilogue: mean heads + bias ----------------
__global__ void k_node2(const float* __restrict__ acc, const float* __restrict__ b2,
                        float* __restrict__ out) {
  int idx = blockIdx.x * blockDim.x + threadIdx.x;
  if (idx >= NN * 32) return;
  const int n = idx >> 5, f = idx & 31;
  out[idx] = 0.5f * (acc[(size_t)n * 64 + f] + acc[(size_t)n * 64 + 32 + f]) + b2[f];
}

// =====================================================================
extern "C" void kernel_launch(void* const* d_in, const int* in_sizes, int n_in,
                              void* d_out, int out_size, void* d_ws, size_t ws_size,
                              hipStream_t stream) {
  (void)in_sizes; (void)n_in; (void)out_size; (void)ws_size;
  const float* x    = (const float*)d_in[0];
  const int*   ei   = (const int*)  d_in[1];
  const float* W1   = (const float*)d_in[2];
  const float* atS  = (const float*)d_in[3];
  const float* atD  = (const float*)d_in[4];
  const float* b1   = (const float*)d_in[5];
  const float* Wl2  = (const float*)d_in[6];
  const float* bl2  = (const float*)d_in[7];
  const float* Wr2  = (const float*)d_in[8];
  const float* br2  = (const float*)d_in[9];
  const float* att2 = (const float*)d_in[10];
  const float* b2   = (const float*)d_in[11];
  float* out = (float*)d_out;

  // workspace layout (floats); everything L2-friendly, layer-2 reuses mx/s/zbuf/acc
  float* ws   = (float*)d_ws;
  float* h    = ws;                          // N*64
  float* as_  = h    + (size_t)NN * 64;      // N*2
  float* ad_  = as_  + (size_t)NN * 2;       // N*2
  float* mx   = ad_  + (size_t)NN * 2;       // N*2
  float* s    = mx   + (size_t)NN * 2;       // N*2
  float* zbuf = s    + (size_t)NN * 2;       // ETOT*2
  float* acc  = zbuf + (size_t)ETOT * 2;     // N*64
  float* h1   = acc  + (size_t)NN * 64;      // N*32
  float* hl   = h1   + (size_t)NN * 32;      // N*64
  float* hr   = hl   + (size_t)NN * 64;      // N*64

  const int T = 256;
  const unsigned gridE  = (ETOT + T - 1) / T;
  const unsigned gridEC = (unsigned)(((size_t)ETOT * 64 + T - 1) / T);

  // ---- Layer 1 (GATConv) ----
  k_gemm1<<<NN / 16, 128, 0, stream>>>(x, W1, h);
  k_attn1<<<(NN * 2 + T - 1) / T, T, 0, stream>>>(h, atS, atD, as_, ad_);
  k_fill<<<(NN * 2 + T - 1) / T, T, 0, stream>>>(mx, NEG_HUGE, NN * 2);
  k_fill<<<(NN * 2 + T - 1) / T, T, 0, stream>>>(s, 0.f, NN * 2);
  k_fill<<<(NN * 64 + T - 1) / T, T, 0, stream>>>(acc, 0.f, NN * 64);
  k_edge1_max<<<gridE, T, 0, stream>>>(ei, as_, ad_, mx);
  k_edge1_z<<<gridE, T, 0, stream>>>(ei, as_, ad_, mx, s, zbuf);
  k_scatter<<<gridEC, T, 0, stream>>>(ei, h, zbuf, s, acc);
  k_node1<<<(NN * 32 + T - 1) / T, T, 0, stream>>>(acc, b1, h1);

  // ---- Layer 2 (GATv2Conv) ----
  k_gemm2<<<NN / 16, 128, 0, stream>>>(h1, Wl2, bl2, Wr2, br2, hl, hr);
  k_fill<<<(NN * 2 + T - 1) / T, T, 0, stream>>>(mx, NEG_HUGE, NN * 2);
  k_fill<<<(NN * 2 + T - 1) / T, T, 0, stream>>>(s, 0.f, NN * 2);
  k_fill<<<(NN * 64 + T - 1) / T, T, 0, stream>>>(acc, 0.f, NN * 64);
  k_edge2_logits<<<(ETOT + 7) / 8, 256, 0, stream>>>(ei, hl, hr, att2, zbuf, mx);
  k_edge2_z<<<gridE, T, 0, stream>>>(ei, zbuf, mx, s);
  k_scatter<<<gridEC, T, 0, stream>>>(ei, hl, zbuf, s, acc);
  k_node2<<<(NN * 32 + T - 1) / T, T, 0, stream>>>(acc, b2, out);
}